// FourierFilter2D_75376676044827
// MI455X (gfx1250) — compile-verified
//
#include <hip/hip_runtime.h>

typedef float v2f __attribute__((ext_vector_type(2)));
typedef float v8f __attribute__((ext_vector_type(8)));

#define BATCH 8
#define HH 64
#define WW 64
#define CC 64
#define FF 64

// One wave (32 lanes) per (h,w) pixel: complex GEMM [8x64]x[64x64] via
// WMMA f32 16x16x4 with 3-multiply (Karatsuba) complex arithmetic:
//   k1 = (Xr+Xi)Wr, k2 = Xr(Wi-Wr), k3 = Xi(Wr+Wi);  re = k1-k3, im = k1+k2
__global__ __launch_bounds__(256, 1) void fourier_filter2d_wmma(
    const float* __restrict__ x_re, const float* __restrict__ x_im,
    const float* __restrict__ w_re, const float* __restrict__ w_im,
    const float* __restrict__ b_re, const float* __restrict__ b_im,
    float* __restrict__ out)
{
    const int lane  = threadIdx.x & 31;
    const int wave  = threadIdx.x >> 5;
    const int pixel = blockIdx.x * 8 + wave;      // 512 blocks * 8 waves = 4096 pixels
    const int h = pixel >> 6;
    const int w = pixel & 63;

    const int l16  = lane & 15;                   // N (f) column / M (b) row within tile
    const int half = lane >> 4;                   // K-half selector per ISA A/B layout

    // ---- accumulators: 4 f-tiles x {t1,t2,t3}; bias folded in: t2=+b_im, t3=-b_re ----
    v8f t1[4], t2[4], t3[4];
#pragma unroll
    for (int ft = 0; ft < 4; ++ft) {
        const float br = b_re[ft * 16 + l16];
        const float bi = b_im[ft * 16 + l16];
#pragma unroll
        for (int r = 0; r < 8; ++r) { t1[ft][r] = 0.0f; t2[ft][r] = bi; t3[ft][r] = -br; }
    }

    const int hw = h * WW + w;
    // A-matrix (X) addressing: lane holds x[b = l16][c = c0 + 2*half + {0,1}]
    const int  b      = l16;
    const bool bvalid = (b < BATCH);
    const int  xbase  = (b * (HH * WW) + hw) * CC + 2 * half;   // + c0 per k-step
    // B-matrix (W) addressing: element idx = c * S + hw*FF + f, S = c-stride
    const int S     = HH * WW * FF;                              // 262144
    const int wbase = (2 * half) * S + hw * FF;                  // + c0*S + ft*16 + l16

    for (int k = 0; k < 16; ++k) {
        const int c0 = 4 * k;

        // ---- A tiles: Xr, Xi (v2f per lane, c-contiguous, 8B aligned); As = Xr+Xi ----
        v2f Ar = {0.0f, 0.0f};
        v2f Ai = {0.0f, 0.0f};
        if (bvalid) {
            Ar = *(const v2f*)(x_re + xbase + c0);
            Ai = *(const v2f*)(x_im + xbase + c0);
        }
        const v2f As = { Ar.x + Ai.x, Ar.y + Ai.y };

        const int wk = wbase + c0 * S;
#pragma unroll
        for (int ft = 0; ft < 4; ++ft) {
            const int wf = wk + ft * 16 + l16;
            // B tile: v.x = row c0+2*half (K=0|2), v.y = row +1 (K=1|3)
            v2f Br, Bi;
            Br.x = w_re[wf];
            Br.y = w_re[wf + S];
            Bi.x = w_im[wf];
            Bi.y = w_im[wf + S];
            const v2f Bd = { Bi.x - Br.x, Bi.y - Br.y };   // Wi - Wr
            const v2f Bs = { Br.x + Bi.x, Br.y + Bi.y };   // Wr + Wi

            // 3 WMMAs instead of 4
            t1[ft] = __builtin_amdgcn_wmma_f32_16x16x4_f32(
                false, As, false, Br, (short)0, t1[ft], false, false);
            t2[ft] = __builtin_amdgcn_wmma_f32_16x16x4_f32(
                false, Ar, false, Bd, (short)0, t2[ft], false, false);
            t3[ft] = __builtin_amdgcn_wmma_f32_16x16x4_f32(
                false, Ai, false, Bs, (short)0, t3[ft], false, false);
        }
    }

    // ---- combine + store: C/D layout VGPR r -> M = r (lanes 0-15); M=8+r is padding ----
    // Output is complex64: interleaved {re, im} float pairs, contiguous in f.
    if (half == 0) {
#pragma unroll
        for (int ft = 0; ft < 4; ++ft) {
#pragma unroll
            for (int r = 0; r < 8; ++r) {
                const int    f    = ft * 16 + l16;
                const size_t oidx = ((size_t)(r * HH + h) * WW + w) * FF + f;
                v2f o = { t1[ft][r] - t3[ft][r],    // re = k1 - k3 (+b_re via t3 init)
                          t1[ft][r] + t2[ft][r] };  // im = k1 + k2 (+b_im via t2 init)
                *(v2f*)(out + 2 * oidx) = o;
            }
        }
    }
}

extern "C" void kernel_launch(void* const* d_in, const int* in_sizes, int n_in,
                              void* d_out, int out_size, void* d_ws, size_t ws_size,
                              hipStream_t stream) {
    (void)in_sizes; (void)n_in; (void)out_size; (void)d_ws; (void)ws_size;
    const float* x_re = (const float*)d_in[0];
    const float* x_im = (const float*)d_in[1];
    const float* w_re = (const float*)d_in[2];
    const float* w_im = (const float*)d_in[3];
    const float* b_re = (const float*)d_in[4];
    const float* b_im = (const float*)d_in[5];
    float* out = (float*)d_out;

    dim3 grid(512);    // 4096 pixels / 8 waves per block
    dim3 block(256);   // 8 wave32 waves
    fourier_filter2d_wmma<<<grid, block, 0, stream>>>(x_re, x_im, w_re, w_im, b_re, b_im, out);
}